// AutoCorrelation_22093311770937
// MI455X (gfx1250) — compile-verified
//
#include <hip/hip_runtime.h>
#include <math.h>

// Problem constants (B,L,D fixed by the reference harness).
#define BB   8
#define LL   1024
#define DD   512
#define TT   2047          // L + S - 1
#define KB   513           // rfft bins of a 1024-pt transform
#define NDIAG 1023         // delta in [-511, 511]
#define TOPK 20            // int(3*log(1024))

typedef float v2f __attribute__((ext_vector_type(2)));
typedef float v8f __attribute__((ext_vector_type(8)));

// ---------------------------------------------------------------------------
// Kernel 1: G[b] = Q[b]^T * K[b]   (512x512 = sum over L of q[l,m]*k[l,n])
// fp32 WMMA 16x16x4, one wave computes a 16(M) x 64(N) tile.
// ---------------------------------------------------------------------------
__global__ __launch_bounds__(128) void gemm_qtk(const float* __restrict__ Q,
                                                const float* __restrict__ Km,
                                                float* __restrict__ G) {
  const int wid  = blockIdx.x * 4 + (threadIdx.x >> 5);  // 0..2047
  const int b    = wid >> 8;                             // 8 batches
  const int t    = wid & 255;                            // 32 mtiles x 8 ntiles
  const int mt   = t >> 3;
  const int nt   = t & 7;
  const int lane = threadIdx.x & 31;
  const int mr   = lane & 15;    // M row (A) / N col (B) within tile
  const int kh   = lane >> 4;    // K half: lanes 0-15 -> K=0,1 ; 16-31 -> K=2,3
  const int mbase = mt * 16;
  const int nbase = nt * 64;

  const float* __restrict__ Qb = Q  + (size_t)b * LL * DD;
  const float* __restrict__ Kb = Km + (size_t)b * LL * DD;

  v8f acc0 = {}, acc1 = {}, acc2 = {}, acc3 = {};

  for (int kk = 0; kk < LL; kk += 4) {
    const int l0 = kk + 2 * kh;
    // A (16x4): A[m][k] = Q[l0+j][mbase+m]
    v2f a;
    a.x = Qb[(size_t)l0 * DD + mbase + mr];
    a.y = Qb[(size_t)(l0 + 1) * DD + mbase + mr];
    // B (4x16): B[k][n] = K[l0+j][nbase+16t+n]
    const float* kr0 = Kb + (size_t)l0 * DD + nbase + mr;
    const float* kr1 = kr0 + DD;
    v2f b0; b0.x = kr0[0];  b0.y = kr1[0];
    v2f b1; b1.x = kr0[16]; b1.y = kr1[16];
    v2f b2; b2.x = kr0[32]; b2.y = kr1[32];
    v2f b3; b3.x = kr0[48]; b3.y = kr1[48];

    acc0 = __builtin_amdgcn_wmma_f32_16x16x4_f32(false, a, false, b0, (short)0, acc0, false, false);
    acc1 = __builtin_amdgcn_wmma_f32_16x16x4_f32(false, a, false, b1, (short)0, acc1, false, false);
    acc2 = __builtin_amdgcn_wmma_f32_16x16x4_f32(false, a, false, b2, (short)0, acc2, false, false);
    acc3 = __builtin_amdgcn_wmma_f32_16x16x4_f32(false, a, false, b3, (short)0, acc3, false, false);
  }

  // C/D layout: VGPR r, lanes 0-15: M=r ; lanes 16-31: M=r+8 ; N = lane&15
  float* __restrict__ Gb = G + (size_t)b * DD * DD;
  const int mo = mbase + (kh ? 8 : 0);
  const int no = nbase + mr;
#pragma unroll
  for (int r = 0; r < 8; ++r) {
    Gb[(size_t)(mo + r) * DD + no +  0] = acc0[r];
    Gb[(size_t)(mo + r) * DD + no + 16] = acc1[r];
    Gb[(size_t)(mo + r) * DD + no + 32] = acc2[r];
    Gb[(size_t)(mo + r) * DD + no + 48] = acc3[r];
  }
}

// ---------------------------------------------------------------------------
// Kernel 2: diagonal sums H[b, delta+511] = sum_{m-n=delta} G[b,m,n]
// ---------------------------------------------------------------------------
__global__ __launch_bounds__(256) void diag_sum(const float* __restrict__ G,
                                                float* __restrict__ H) {
  const int idx = blockIdx.x * 256 + threadIdx.x;
  if (idx >= BB * NDIAG) return;
  const int b     = idx / NDIAG;
  const int delta = (idx % NDIAG) - 511;
  const float* __restrict__ Gb = G + (size_t)b * DD * DD;
  const int mlo = delta > 0 ? delta : 0;
  const int mhi = delta > 0 ? (DD - 1) : (DD - 1 + delta);
  float s = 0.f;
  for (int m = mlo; m <= mhi; ++m) s += Gb[(size_t)m * DD + (m - delta)];
  H[idx] = s;
}

// ---------------------------------------------------------------------------
// Kernel 3: Pm[b,k] = sum_delta H[b,delta] * e^{-2*pi*i*k*delta/1024}
// Integer phase reduction keeps sincos args in [0, 2*pi).
// ---------------------------------------------------------------------------
__global__ __launch_bounds__(256) void dft1(const float* __restrict__ H,
                                            float* __restrict__ PmRe,
                                            float* __restrict__ PmIm) {
  const int idx = blockIdx.x * 256 + threadIdx.x;
  if (idx >= BB * KB) return;
  const int b = idx / KB;
  const int k = idx % KB;
  const float* __restrict__ Hb = H + b * NDIAG;
  const float w = 6.28318530717958647692f / 1024.0f;
  float re = 0.f, im = 0.f;
  for (int dlt = -511; dlt <= 511; ++dlt) {
    const float h = Hb[dlt + 511];
    int ph = (k * dlt) % 1024;           // exact integer phase
    if (ph < 0) ph += 1024;
    float s, c;
    sincosf(w * (float)ph, &s, &c);
    re += h * c;
    im -= h * s;
  }
  PmRe[idx] = re;
  PmIm[idx] = im;
}

// ---------------------------------------------------------------------------
// Kernel 4: mean_value[b,t] = irfft-to-2047 of Pm, scaled by 1/(2047*1024)
// ---------------------------------------------------------------------------
__global__ __launch_bounds__(256) void dft2(const float* __restrict__ PmRe,
                                            const float* __restrict__ PmIm,
                                            float* __restrict__ MV) {
  const int idx = blockIdx.x * 256 + threadIdx.x;
  if (idx >= BB * TT) return;
  const int b = idx / TT;
  const int t = idx % TT;
  const float* __restrict__ Rb = PmRe + b * KB;
  const float* __restrict__ Ib = PmIm + b * KB;
  const float w = 6.28318530717958647692f / 2047.0f;
  float acc = Rb[0];                     // k=0 term (imag ignored, as irfft does)
  for (int k = 1; k < KB; ++k) {
    const int ph = (k * t) % 2047;       // exact integer phase (fits in int)
    float s, c;
    sincosf(w * (float)ph, &s, &c);
    acc += 2.0f * (Rb[k] * c - Ib[k] * s);
  }
  MV[idx] = acc * (1.0f / (2047.0f * 1024.0f));
}

// ---------------------------------------------------------------------------
// Kernel 5: per-batch top-20 (value desc, tie -> lower index, like lax.top_k),
// softmax -> weights; batch 0 additionally publishes its 20 indices (shifts).
// One block per batch.
// ---------------------------------------------------------------------------
__global__ __launch_bounds__(256) void topk_softmax(const float* __restrict__ MV,
                                                    float* __restrict__ Wout,
                                                    int* __restrict__ shifts) {
  __shared__ float svals[TT];
  __shared__ float rv[256];
  __shared__ int   ri[256];
  __shared__ float selV[TOPK];
  __shared__ int   selI[TOPK];

  const int b = blockIdx.x;
  const int tid = threadIdx.x;
  for (int t = tid; t < TT; t += 256) svals[t] = MV[b * TT + t];
  __syncthreads();

  for (int r = 0; r < TOPK; ++r) {
    float bestv = -INFINITY;
    int   besti = 0x7fffffff;
    for (int t = tid; t < TT; t += 256) {
      const float v = svals[t];
      if (v > bestv || (v == bestv && t < besti)) { bestv = v; besti = t; }
    }
    rv[tid] = bestv; ri[tid] = besti;
    __syncthreads();
    for (int off = 128; off > 0; off >>= 1) {
      if (tid < off) {
        const float v2 = rv[tid + off];
        const int   i2 = ri[tid + off];
        if (v2 > rv[tid] || (v2 == rv[tid] && i2 < ri[tid])) { rv[tid] = v2; ri[tid] = i2; }
      }
      __syncthreads();
    }
    if (tid == 0) {
      selV[r] = rv[0];
      selI[r] = ri[0];
      svals[ri[0]] = -INFINITY;          // remove winner for next round
    }
    __syncthreads();
  }

  if (tid == 0) {
    const float mx = selV[0];            // sorted descending -> max is first
    float e[TOPK];
    float sum = 0.f;
    for (int r = 0; r < TOPK; ++r) { e[r] = expf(selV[r] - mx); sum += e[r]; }
    const float inv = 1.0f / sum;
    for (int r = 0; r < TOPK; ++r) Wout[b * TOPK + r] = e[r] * inv;
    if (b == 0) {
      for (int r = 0; r < TOPK; ++r) shifts[r] = selI[r];
    }
  }
}

// ---------------------------------------------------------------------------
// Kernel 6: out[b,l,d] = sum_k w[b,k] * V[b, (l + s_k) mod 1024, d]
// ---------------------------------------------------------------------------
__global__ __launch_bounds__(256) void apply_agg(const float* __restrict__ V,
                                                 const float* __restrict__ W,
                                                 const int* __restrict__ shifts,
                                                 float* __restrict__ out) {
  const int idx = blockIdx.x * 256 + threadIdx.x;   // BB*LL*DD total
  const int d = idx & (DD - 1);
  const int l = (idx >> 9) & (LL - 1);
  const int b = idx >> 19;
  const float* __restrict__ Vb = V + ((size_t)b << 19);
  const float* __restrict__ Wb = W + b * TOPK;
  float acc = 0.f;
#pragma unroll
  for (int k = 0; k < TOPK; ++k) {
    const int row = (l + shifts[k]) & (LL - 1);
    acc += Wb[k] * Vb[((size_t)row << 9) + d];
  }
  out[idx] = acc;
}

// ---------------------------------------------------------------------------
// Launcher. Workspace layout (floats): G[8*512*512] | H[8*1023] |
// PmRe[8*513] | PmIm[8*513] | MV[8*2047] | W[8*20] | shifts[20 ints]
// (~8.2 MB total). All buffers fully written before being read -> no init
// needed, fully deterministic across replays.
// ---------------------------------------------------------------------------
extern "C" void kernel_launch(void* const* d_in, const int* in_sizes, int n_in,
                              void* d_out, int out_size, void* d_ws, size_t ws_size,
                              hipStream_t stream) {
  const float* Q  = (const float*)d_in[0];
  const float* Kk = (const float*)d_in[1];
  const float* V  = (const float*)d_in[2];

  float* ws    = (float*)d_ws;
  float* G     = ws;                         // 8*512*512
  float* H     = G + (size_t)BB * DD * DD;   // 8*1023
  float* PmRe  = H + BB * NDIAG;             // 8*513
  float* PmIm  = PmRe + BB * KB;             // 8*513
  float* MV    = PmIm + BB * KB;             // 8*2047
  float* Wgt   = MV + BB * TT;               // 8*20
  int*   shifts = (int*)(Wgt + BB * TOPK);   // 20

  // 2048 wave-tiles (8 batches x 32 mtiles x 8 ntiles), 4 waves / block
  gemm_qtk<<<512, 128, 0, stream>>>(Q, Kk, G);
  diag_sum<<<(BB * NDIAG + 255) / 256, 256, 0, stream>>>(G, H);
  dft1<<<(BB * KB + 255) / 256, 256, 0, stream>>>(H, PmRe, PmIm);
  dft2<<<(BB * TT + 255) / 256, 256, 0, stream>>>(PmRe, PmIm, MV);
  topk_softmax<<<BB, 256, 0, stream>>>(MV, Wgt, shifts);
  apply_agg<<<(BB * LL * DD) / 256, 256, 0, stream>>>(V, Wgt, shifts, (float*)d_out);
}